// MAGNN_attn_intra_5308579578456
// MI455X (gfx1250) — compile-verified
//
#include <hip/hip_runtime.h>
#include <math.h>

#define HEADS 8
#define HD 256          // H*D
#define NEG_SLOPE 0.01f
#define FLT_NEG_MAX -3.402823466e+38f

typedef float v2f __attribute__((ext_vector_type(2)));
typedef float v8f __attribute__((ext_vector_type(8)));

// ---------------- init: zero out + s, set m = -FLT_MAX ----------------
__global__ void magnn_init(float* __restrict__ mMax, float* __restrict__ sSum,
                           float* __restrict__ out, long nOut, long nNH) {
    long i = (long)blockIdx.x * blockDim.x + threadIdx.x;
    if (i < nOut) out[i] = 0.f;
    if (i < nNH) { mMax[i] = FLT_NEG_MAX; sSum[i] = 0.f; }
}

// ---------------- pass A: e = leakyrelu(feat . attn_r) via f32 WMMA ----------------
// One wave per 16-edge tile. D = A(16x4 feat slab) x B(4x16 block-diag attn) + C,
// chained 64x over K=0..255. C/D layout: lane -> N(head), VGPR v -> M (edge v + 8*half).
__global__ void magnn_pass_a(const float* __restrict__ feat,
                             const float* __restrict__ attn,
                             const int*   __restrict__ mp,
                             float* __restrict__ eBuf,
                             float* __restrict__ mMax,
                             int E) {
    int wave   = blockIdx.x * (blockDim.x >> 5) + (threadIdx.x >> 5);
    int nTiles = (E + 15) >> 4;
    if (wave >= nTiles) return;              // wave-uniform: EXEC stays all-ones for WMMA

    int lane = threadIdx.x & 31;
    int half = lane >> 4;                    // 0: K={0,1}, 1: K={2,3} (A layout)
    int m    = lane & 15;                    // A: row M; B/C/D: column N

    long tileBase = (long)wave * 16;
    long rowEdge  = tileBase + m;
    if (rowEdge >= E) rowEdge = E - 1;       // clamp (stores guarded below)
    const float* arow = feat + rowEdge * (long)HD;

    v8f c = {0.f, 0.f, 0.f, 0.f, 0.f, 0.f, 0.f, 0.f};

    #pragma unroll
    for (int k0 = 0; k0 < HD; k0 += 4) {
        int kk = k0 + 2 * half;
        v2f a, b;
        a.x = arow[kk];
        a.y = arow[kk + 1];
        int h = k0 >> 5;                     // only head column h nonzero in this K-block
        float bx = attn[kk];                 // attn flat index h*32 + d == k0 + (kk-k0)
        float by = attn[kk + 1];
        b.x = (m == h) ? bx : 0.f;
        b.y = (m == h) ? by : 0.f;
        // 8 args: (neg_a, A, neg_b, B, c_mod, C, reuse_a, reuse_b)
        c = __builtin_amdgcn_wmma_f32_16x16x4_f32(false, a, false, b,
                                                  (short)0, c, false, false);
    }

    if (m < HEADS) {
        #pragma unroll
        for (int v = 0; v < 8; ++v) {
            long eIdx = tileBase + v + 8 * half;   // M = v + 8*half
            if (eIdx < E) {
                float ev = c[v];
                ev = (ev >= 0.f) ? ev : NEG_SLOPE * ev;   // leaky relu
                eBuf[eIdx * HEADS + m] = ev;
                int node = mp[eIdx * 2];                   // dst = metapath_idx[:,0]
                atomicMax(&mMax[(long)node * HEADS + m], ev);
            }
        }
    }
}

// ---------------- pass B: ex = exp(e - m[dst]); s[dst] += ex ----------------
__global__ void magnn_pass_b(const int* __restrict__ mp,
                             const float* __restrict__ mMax,
                             float* __restrict__ eBuf,
                             float* __restrict__ sSum,
                             long EH) {
    long i = (long)blockIdx.x * blockDim.x + threadIdx.x;
    if (i >= EH) return;
    long edge = i >> 3;
    int  h    = (int)(i & 7);
    int  node = mp[edge * 2];
    float ex = expf(eBuf[i] - mMax[(long)node * HEADS + h]);
    eBuf[i] = ex;
    atomicAdd(&sSum[(long)node * HEADS + h], ex);
}

// ---------------- pass C: out[dst] += feat * ex / s[dst] ----------------
// One wave per edge; lane covers 8 consecutive features (two float4 loads).
__global__ void magnn_pass_c(const float* __restrict__ feat,
                             const int*   __restrict__ mp,
                             const float* __restrict__ exBuf,
                             const float* __restrict__ sSum,
                             float* __restrict__ out,
                             int E) {
    int edge = blockIdx.x * (blockDim.x >> 5) + (threadIdx.x >> 5);
    if (edge >= E) return;
    int lane = threadIdx.x & 31;
    int node = mp[(long)edge * 2];
    int h    = lane >> 2;                    // 8 features -> head = lane/4
    float a  = exBuf[(long)edge * HEADS + h] / sSum[(long)node * HEADS + h];

    const float4* frow = (const float4*)(feat + (long)edge * HD) + lane * 2;
    float4 f0 = frow[0];
    float4 f1 = frow[1];
    float* orow = out + (long)node * HD + lane * 8;
    atomicAdd(orow + 0, f0.x * a);
    atomicAdd(orow + 1, f0.y * a);
    atomicAdd(orow + 2, f0.z * a);
    atomicAdd(orow + 3, f0.w * a);
    atomicAdd(orow + 4, f1.x * a);
    atomicAdd(orow + 5, f1.y * a);
    atomicAdd(orow + 6, f1.z * a);
    atomicAdd(orow + 7, f1.w * a);
}

// ---------------- pass D: out = elu(out) (already divided by s in pass C) ----------------
__global__ void magnn_pass_d(float* __restrict__ out, long nOut) {
    long i = (long)blockIdx.x * blockDim.x + threadIdx.x;
    if (i >= nOut) return;
    float v = out[i];
    out[i] = (v > 0.f) ? v : expm1f(v);
}

extern "C" void kernel_launch(void* const* d_in, const int* in_sizes, int n_in,
                              void* d_out, int out_size, void* d_ws, size_t ws_size,
                              hipStream_t stream) {
    const float* feat = (const float*)d_in[0];   // (E, 256) f32
    const float* attn = (const float*)d_in[1];   // (1, 8, 32) f32
    const int*   mp   = (const int*)d_in[2];     // (E, 2) i32
    // d_in[3] = num_nodes (device scalar); derive N from out_size instead.
    int E = in_sizes[0] / HD;
    int N = out_size / HD;
    float* out = (float*)d_out;

    float* ws   = (float*)d_ws;
    float* eBuf = ws;                                  // E*8 floats (e, then ex)
    float* mMax = eBuf + (size_t)E * HEADS;            // N*8 floats
    float* sSum = mMax + (size_t)N * HEADS;            // N*8 floats

    long nOut = (long)N * HD;
    long nNH  = (long)N * HEADS;
    long EH   = (long)E * HEADS;

    magnn_init<<<(unsigned)((nOut + 255) / 256), 256, 0, stream>>>(mMax, sSum, out, nOut, nNH);

    int tiles = (E + 15) / 16;                         // waves; 8 waves per 256-thread block
    magnn_pass_a<<<(unsigned)((tiles + 7) / 8), 256, 0, stream>>>(feat, attn, mp, eBuf, mMax, E);

    magnn_pass_b<<<(unsigned)((EH + 255) / 256), 256, 0, stream>>>(mp, mMax, eBuf, sSum, EH);

    magnn_pass_c<<<(unsigned)((E + 7) / 8), 256, 0, stream>>>(feat, mp, eBuf, sSum, out, E);

    magnn_pass_d<<<(unsigned)((nOut + 255) / 256), 256, 0, stream>>>(out, nOut);
}